// MatrixExpander_5308579578041
// MI455X (gfx1250) — compile-verified
//
#include <hip/hip_runtime.h>
#include <stdint.h>

// MatrixExpander: out[b,c, r, col] = A[b,c, r/8, col/8]  (kron(A, ones(8,8)))
// B=C=16, Ar=Ac=64, mp=8  ->  A: 1,048,576 f32 (4 MB), out: 67,108,864 f32 (256 MB)
//
// Store-bandwidth bound (~11 us at 23.3 TB/s). Zero required FLOPs, so no WMMA;
// we use the CDNA5 async data path (global_load_async_to_lds_b32 + s_wait_asynccnt,
// ASYNCcnt machinery) for input staging and 16B non-temporal global stores for output.

typedef float v4f __attribute__((ext_vector_type(4)));

__global__ __launch_bounds__(256) void MatrixExpander_kernel(
    const float* __restrict__ A, float* __restrict__ out)
{
    __shared__ float tileA[256];

    const unsigned tid = threadIdx.x;
    const unsigned g   = blockIdx.x * 256u + tid;   // flat A element index, < 2^20

    // ---- CDNA5 async copy: this block's 256 A floats -> LDS ----
    // GVS addressing: mem = SADDR(64b sgpr pair) + VADDR(32b byte offset)
    // LDS dest VGPR holds the byte offset within the workgroup LDS allocation;
    // the low 32 bits of a flat pointer to a __shared__ object are exactly that
    // offset (the ISA uses the low 32 address bits).  Passing &tileA[tid] into the asm also
    // makes tileA escape, so the compiler keeps the LDS allocation and the loads.
    {
        unsigned lds_addr  = (unsigned)(uintptr_t)(&tileA[tid]);
        unsigned glob_off  = g * 4u;
        asm volatile("global_load_async_to_lds_b32 %0, %1, %2"
                     :
                     : "v"(lds_addr), "v"(glob_off), "s"(A)
                     : "memory");
        asm volatile("s_wait_asynccnt 0" ::: "memory");
    }
    __syncthreads();

    const float a = tileA[tid];
    const v4f val = {a, a, a, a};

    // g = ((bc*64 + ar)*64 + ac); output base = bc*262144 + ar*4096 + ac*8
    //   = ((g >> 6) << 12) + ((g & 63) << 3)
    const unsigned long long obase =
        ((unsigned long long)(g >> 6) << 12) + (unsigned long long)((g & 63u) << 3);

    float* p = out + obase;
#pragma unroll
    for (int r = 0; r < 8; ++r) {
        // two 16B non-temporal stores cover this thread's 8 output columns
        __builtin_nontemporal_store(val, (v4f*)(p));
        __builtin_nontemporal_store(val, (v4f*)(p + 4));
        p += 512;   // next output row (Ac * mp)
    }
}

extern "C" void kernel_launch(void* const* d_in, const int* in_sizes, int n_in,
                              void* d_out, int out_size, void* d_ws, size_t ws_size,
                              hipStream_t stream) {
    (void)in_sizes; (void)n_in; (void)d_ws; (void)ws_size; (void)out_size;

    const float* A = (const float*)d_in[0];   // (16,16,64,64) f32
    // d_in[1] = left_mat, d_in[2] = right_mat: fixed 0/1 selection matrices;
    // their product is exactly the 8x8 block broadcast, so they are not needed.
    float* out = (float*)d_out;               // (16,16,512,512) f32

    // 1,048,576 A elements / 256 threads = 4096 blocks; one 8x8 block per thread
    MatrixExpander_kernel<<<dim3(4096), dim3(256), 0, stream>>>(A, out);
}